// moleculeGCN_90099823935877
// MI455X (gfx1250) — compile-verified
//
#include <hip/hip_runtime.h>
#include <hip/hip_bf16.h>

// ---------------------------------------------------------------------------
// 3-layer GCN for MI455X (gfx1250, wave32).
//  - Dense x@W via V_WMMA_F32_16X16X4_F32 (full f32 precision).
//  - Edge scatter-add via global f32 atomics (L2-resident working set).
//  - dis = rsqrt(deg) and edge norms precomputed once (layer-invariant).
//  - ReLU is a compile-time template flag (no per-element cndmask in K-loop).
// ---------------------------------------------------------------------------

typedef __attribute__((ext_vector_type(2))) float v2f;
typedef __attribute__((ext_vector_type(8))) float v8f;

#define DFEAT 64

// ---------------------------- small utility kernels ------------------------

__global__ void fill_kernel(float* __restrict__ p, float v, long n) {
    long i = (long)blockIdx.x * blockDim.x + threadIdx.x;
    if (i < n) p[i] = v;
}

__global__ void deg_kernel(const int* __restrict__ dst, float* __restrict__ deg, int E) {
    int e = blockIdx.x * blockDim.x + threadIdx.x;
    if (e < E) atomicAdd(&deg[dst[e]], 1.0f);
}

__global__ void rsqrt_kernel(float* __restrict__ d, int n) {
    int i = blockIdx.x * blockDim.x + threadIdx.x;
    if (i < n) d[i] = rsqrtf(d[i]);
}

__global__ void norm_kernel(const int* __restrict__ src, const int* __restrict__ dst,
                            const float* __restrict__ dis, float* __restrict__ nrm, int E) {
    int e = blockIdx.x * blockDim.x + threadIdx.x;
    if (e < E) nrm[e] = dis[src[e]] * dis[dst[e]];
}

// ------------------------------- WMMA GEMM ---------------------------------
// One wave32 computes a 16(M) x 64(N) tile of h = relu?(in) @ W.
// Fused store:  h -> hbuf,   agg_init = h*dis(row)^2 + b -> aggbuf.
// A(16x4) lane layout: M = lane%16, K = vcomp + 2*(lane/16)
// B(4x16) lane layout: N = lane%16, K = vcomp + 2*(lane/16)
// C/D   lane layout:   M = r + 8*(lane/16), N = lane%16   (r = vgpr 0..7)

template <bool RELU_IN>
__global__ __launch_bounds__(256)
void gcn_gemm_kernel(const float* in, const float* __restrict__ W,
                     const float* __restrict__ b, const float* __restrict__ dis,
                     float* h, float* agg, int n)
{
    const int wave = threadIdx.x >> 5;
    const int lane = threadIdx.x & 31;
    const int row0 = (blockIdx.x * 8 + wave) * 16;
    if (row0 >= n) return;                 // wave-uniform: EXEC stays all-ones

    const int lm = lane & 15;              // M / N within tile
    const int lh = lane >> 4;              // K-pair selector

    const float* arow = in + (size_t)(row0 + lm) * DFEAT + 2 * lh;

    v8f acc0 = {}, acc1 = {}, acc2 = {}, acc3 = {};

#pragma unroll
    for (int s = 0; s < 16; ++s) {
        const int k0 = 4 * s;
        float a0 = arow[k0];
        float a1 = arow[k0 + 1];
        if (RELU_IN) { a0 = fmaxf(a0, 0.0f); a1 = fmaxf(a1, 0.0f); }
        v2f A; A[0] = a0; A[1] = a1;

        const float* wr0 = W + (size_t)(k0 + 2 * lh) * DFEAT + lm;  // K = k0+2*lh
        const float* wr1 = wr0 + DFEAT;                             // K = k0+2*lh+1

        v2f B0; B0[0] = wr0[0];  B0[1] = wr1[0];
        v2f B1; B1[0] = wr0[16]; B1[1] = wr1[16];
        v2f B2; B2[0] = wr0[32]; B2[1] = wr1[32];
        v2f B3; B3[0] = wr0[48]; B3[1] = wr1[48];

        acc0 = __builtin_amdgcn_wmma_f32_16x16x4_f32(false, A, false, B0, (short)0, acc0, false, false);
        acc1 = __builtin_amdgcn_wmma_f32_16x16x4_f32(false, A, false, B1, (short)0, acc1, false, false);
        acc2 = __builtin_amdgcn_wmma_f32_16x16x4_f32(false, A, false, B2, (short)0, acc2, false, false);
        acc3 = __builtin_amdgcn_wmma_f32_16x16x4_f32(false, A, false, B3, (short)0, acc3, false, false);
    }

#define STORE_TILE(ACC, T)                                      \
    {                                                           \
        _Pragma("unroll")                                       \
        for (int r = 0; r < 8; ++r) {                           \
            const int row = row0 + r + 8 * lh;                  \
            const int col = (T) * 16 + lm;                      \
            const float val = (ACC)[r];                         \
            const size_t idx = (size_t)row * DFEAT + col;       \
            h[idx] = val;                                       \
            const float dd = dis[row];                          \
            agg[idx] = fmaf(val, dd * dd, b[col]);              \
        }                                                       \
    }
    STORE_TILE(acc0, 0)
    STORE_TILE(acc1, 1)
    STORE_TILE(acc2, 2)
    STORE_TILE(acc3, 3)
#undef STORE_TILE
}

// ------------------------------ edge scatter -------------------------------
// One thread per (edge, 4 features): agg[dst] += h[src] * norm[e]

__global__ void gcn_scatter_kernel(const int* __restrict__ src, const int* __restrict__ dst,
                                   const float* __restrict__ nrm, const float* __restrict__ h,
                                   float* agg, int E)
{
    long tid = (long)blockIdx.x * blockDim.x + threadIdx.x;
    if (tid >= (long)E * 16) return;
    const int e  = (int)(tid >> 4);
    const int f4 = (int)(tid & 15) * 4;
    const int s  = src[e];
    const int d  = dst[e];
    const float w = nrm[e];
    const float4 v = *reinterpret_cast<const float4*>(h + (size_t)s * DFEAT + f4);
    float* ap = agg + (size_t)d * DFEAT + f4;
    atomicAdd(ap + 0, v.x * w);
    atomicAdd(ap + 1, v.y * w);
    atomicAdd(ap + 2, v.z * w);
    atomicAdd(ap + 3, v.w * w);
}

// ------------------------------- mean pool ---------------------------------

__global__ void pool_accum_kernel(const float* __restrict__ h, const int* __restrict__ batch,
                                  float* out, float* cnt, int n)
{
    long tid = (long)blockIdx.x * blockDim.x + threadIdx.x;
    if (tid >= (long)n * 16) return;
    const int node = (int)(tid >> 4);
    const int f4   = (int)(tid & 15) * 4;
    const int g    = batch[node] - batch[0];
    const float4 v = *reinterpret_cast<const float4*>(h + (size_t)node * DFEAT + f4);
    float* op = out + (size_t)g * DFEAT + f4;
    atomicAdd(op + 0, v.x);
    atomicAdd(op + 1, v.y);
    atomicAdd(op + 2, v.z);
    atomicAdd(op + 3, v.w);
    if (f4 == 0) atomicAdd(&cnt[g], 1.0f);
}

__global__ void pool_div_kernel(float* out, const float* __restrict__ cnt, int ngraphs) {
    int i = blockIdx.x * blockDim.x + threadIdx.x;
    if (i < ngraphs * DFEAT) out[i] /= fmaxf(cnt[i >> 6], 1.0f);
}

// ------------------------------- launcher ----------------------------------

extern "C" void kernel_launch(void* const* d_in, const int* in_sizes, int n_in,
                              void* d_out, int out_size, void* d_ws, size_t ws_size,
                              hipStream_t stream) {
    const float* x     = (const float*)d_in[0];
    const int*   ei    = (const int*)d_in[1];   // [2*E] : src then dst
    // d_in[2] = edge_attr (unused by reference)
    const int*   batch = (const int*)d_in[3];
    const float* W1 = (const float*)d_in[4]; const float* b1 = (const float*)d_in[5];
    const float* W2 = (const float*)d_in[6]; const float* b2 = (const float*)d_in[7];
    const float* W3 = (const float*)d_in[8]; const float* b3 = (const float*)d_in[9];

    const int N = in_sizes[0] / DFEAT;      // 50000 (divisible by 16)
    const int E = in_sizes[1] / 2;          // 800000
    const int G = out_size / DFEAT;         // 1024
    const int* src = ei;
    const int* dst = ei + E;

    // workspace carve-out (~29.1 MB total)
    char*  ws  = (char*)d_ws;
    size_t off = 0;
    auto take = [&](size_t bytes) -> float* {
        float* p = (float*)(ws + off);
        off = (off + bytes + 255) & ~(size_t)255;
        return p;
    };
    float* dis  = take((size_t)N * 4);          // degree -> rsqrt(degree)
    float* nrm  = take((size_t)E * 4);          // per-edge norm
    float* cnt  = take((size_t)G * 4);          // pool counts
    float* bufH = take((size_t)N * DFEAT * 4);  // h = in @ W
    float* bufA = take((size_t)N * DFEAT * 4);  // aggregated output
    (void)ws_size; (void)n_in;
    float* out = (float*)d_out;

    const int B = 256;
    const int tiles     = (N + 15) / 16;
    const int gemm_grid = (tiles + 7) / 8;
    const long sthreads = (long)E * 16;
    const long pthreads = (long)N * 16;

    // normalization terms (once; shared by all 3 layers)
    fill_kernel <<<(N + B - 1) / B, B, 0, stream>>>(dis, 1.0f, N);
    deg_kernel  <<<(E + B - 1) / B, B, 0, stream>>>(dst, dis, E);
    rsqrt_kernel<<<(N + B - 1) / B, B, 0, stream>>>(dis, N);
    norm_kernel <<<(E + B - 1) / B, B, 0, stream>>>(src, dst, dis, nrm, E);

    // layer 1: in = x
    gcn_gemm_kernel<false><<<gemm_grid, 256, 0, stream>>>(x, W1, b1, dis, bufH, bufA, N);
    gcn_scatter_kernel<<<(int)((sthreads + B - 1) / B), B, 0, stream>>>(src, dst, nrm, bufH, bufA, E);

    // layer 2: in = relu(bufA)  (each wave reads its rows fully before writing them)
    gcn_gemm_kernel<true><<<gemm_grid, 256, 0, stream>>>(bufA, W2, b2, dis, bufH, bufA, N);
    gcn_scatter_kernel<<<(int)((sthreads + B - 1) / B), B, 0, stream>>>(src, dst, nrm, bufH, bufA, E);

    // layer 3: in = relu(bufA), no relu on output
    gcn_gemm_kernel<true><<<gemm_grid, 256, 0, stream>>>(bufA, W3, b3, dis, bufH, bufA, N);
    gcn_scatter_kernel<<<(int)((sthreads + B - 1) / B), B, 0, stream>>>(src, dst, nrm, bufH, bufA, E);

    // global mean pool
    fill_kernel      <<<(G * DFEAT + B - 1) / B, B, 0, stream>>>(out, 0.0f, (long)G * DFEAT);
    fill_kernel      <<<(G + B - 1) / B, B, 0, stream>>>(cnt, 0.0f, G);
    pool_accum_kernel<<<(int)((pthreads + B - 1) / B), B, 0, stream>>>(bufA, batch, out, cnt, N);
    pool_div_kernel  <<<(G * DFEAT + B - 1) / B, B, 0, stream>>>(out, cnt, G);
}